// LSTM_1365799600437
// MI455X (gfx1250) — compile-verified
//
#include <hip/hip_runtime.h>
#include <math.h>

// LSTM + attention pooling for MI455X (gfx1250, wave32).
// B=64, T=1024, I=256, H=512, 4H=2048. fp32 via V_WMMA_F32_16X16X4_F32.
// Persistent kernel: the 1024-step recurrence runs inside ONE launch with a
// device-wide atomic split-barrier between steps (launch latency >> step cost).

typedef __attribute__((ext_vector_type(2))) float v2f;
typedef __attribute__((ext_vector_type(8))) float v8f;

#define B_ 64
#define T_ 1024
#define I_ 256
#define H_ 512
#define NBLK 160  // 128 cell blocks + 32 attention blocks, 128 thr (4 waves) each

__device__ __forceinline__ v8f wmma4(v2f a, v2f b, v8f c) {
  return __builtin_amdgcn_wmma_f32_16x16x4_f32(
      /*neg_a=*/false, a, /*neg_b=*/false, b,
      /*c_mod=*/(short)0, c, /*reuse_a=*/false, /*reuse_b=*/false);
}

__device__ __forceinline__ float sigmoidf_(float x) {
  return 1.0f / (1.0f + __expf(-x));
}

// Device-wide split barrier: acq-rel counter + generation flag (agent scope).
__device__ __forceinline__ void grid_barrier(unsigned* cnt, unsigned* gen) {
  __syncthreads();
  if (threadIdx.x == 0) {
    unsigned g =
        __hip_atomic_load(gen, __ATOMIC_RELAXED, __HIP_MEMORY_SCOPE_AGENT);
    unsigned old = __hip_atomic_fetch_add(cnt, 1u, __ATOMIC_ACQ_REL,
                                          __HIP_MEMORY_SCOPE_AGENT);
    if (old == (unsigned)(NBLK - 1)) {
      __hip_atomic_store(cnt, 0u, __ATOMIC_RELAXED, __HIP_MEMORY_SCOPE_AGENT);
      __hip_atomic_store(gen, g + 1u, __ATOMIC_RELEASE,
                         __HIP_MEMORY_SCOPE_AGENT);
    } else {
      while (__hip_atomic_load(gen, __ATOMIC_ACQUIRE,
                               __HIP_MEMORY_SCOPE_AGENT) == g) {
        __builtin_amdgcn_s_sleep(2);
      }
    }
  }
  __syncthreads();
}

__global__ void lstm_init_kernel(float* h0, float* h1, float* c, float* mb,
                                 float* sb, float* wb, unsigned* bar) {
  int i = blockIdx.x * blockDim.x + threadIdx.x;
  if (i < B_ * H_) {
    h0[i] = 0.0f;
    h1[i] = 0.0f;
    c[i] = 0.0f;
    mb[i] = -INFINITY;
    sb[i] = 0.0f;
    wb[i] = 0.0f;
  }
  if (i < 2) bar[i] = 0u;  // counter, generation
}

// Persistent kernel: per barrier interval t, blocks 0..127 compute the LSTM
// cell for step t (h_t from h_{t-1}); blocks 128..159 run the attention GEMM
// + online-softmax update for h_{t-1}. Both only read h_{t-1} (ping-pong).
__global__ __launch_bounds__(128) void lstm_persist_kernel(
    const float* __restrict__ x, const float* __restrict__ W_ih,
    const float* __restrict__ W_hh, const float* __restrict__ b_ih,
    const float* __restrict__ b_hh, const float* __restrict__ W_attn,
    const float* __restrict__ b_attn, float* __restrict__ hb0,
    float* __restrict__ hb1, float* __restrict__ cbuf,
    float* __restrict__ mbuf, float* __restrict__ sbuf,
    float* __restrict__ wbuf, float* __restrict__ out, unsigned* bar) {
  __shared__ float lds[4][16][16];
  unsigned* cnt = bar;
  unsigned* gen = bar + 1;

  const int lane = threadIdx.x & 31;
  const int w = threadIdx.x >> 5;  // wave in block (0..3)
  const int l16 = lane & 15;
  const int lh = lane >> 4;        // lane half -> K sub-pair of the 16x16x4 frag

  const bool is_cell = (blockIdx.x < 128);

  for (int t = 0; t <= T_; ++t) {
    const float* hp = (t & 1) ? hb0 : hb1;  // h_{t-1}
    float* hc = (t & 1) ? hb1 : hb0;        // h_t

    if (is_cell) {
      if (t < T_) {
        const int mblk = blockIdx.x >> 5;    // 0..3 : batch block of 16
        const int nstrip = blockIdx.x & 31;  // 0..31 : 16-wide strip per gate
        const int m_base = mblk * 16;
        // wave w owns gate chunk w (i,f,g,o); W row = gate column in [0,2048)
        const int n_row = w * H_ + nstrip * 16 + l16;

        v8f acc = {0.f, 0.f, 0.f, 0.f, 0.f, 0.f, 0.f, 0.f};
        {  // x_t @ W_ih^T (K = 256)
          const float* ap =
              x + (size_t)(m_base + l16) * (T_ * I_) + (size_t)t * I_ + 2 * lh;
          const float* bp = W_ih + (size_t)n_row * I_ + 2 * lh;
#pragma unroll 8
          for (int k = 0; k < I_; k += 4) {
            acc = wmma4(*(const v2f*)(ap + k), *(const v2f*)(bp + k), acc);
          }
        }
        {  // h_{t-1} @ W_hh^T (K = 512)
          const float* ap = hp + (size_t)(m_base + l16) * H_ + 2 * lh;
          const float* bp = W_hh + (size_t)n_row * H_ + 2 * lh;
#pragma unroll 8
          for (int k = 0; k < H_; k += 4) {
            acc = wmma4(*(const v2f*)(ap + k), *(const v2f*)(bp + k), acc);
          }
        }
        const float biasv = b_ih[n_row] + b_hh[n_row];
#pragma unroll
        for (int r = 0; r < 8; ++r) lds[w][lh * 8 + r][l16] = acc[r] + biasv;
        __syncthreads();

        const int hcol_base = nstrip * 16;
#pragma unroll
        for (int j = 0; j < 2; ++j) {
          int e = threadIdx.x * 2 + j;  // 0..255 over the 16x16 (b,h) tile
          int m = e >> 4;
          int n = e & 15;
          float iv = sigmoidf_(lds[0][m][n]);
          float fv = sigmoidf_(lds[1][m][n]);
          float gv = tanhf(lds[2][m][n]);
          float ov = sigmoidf_(lds[3][m][n]);
          size_t idx = (size_t)(m_base + m) * H_ + (hcol_base + n);
          float cn = fv * cbuf[idx] + iv * gv;
          cbuf[idx] = cn;
          hc[idx] = ov * tanhf(cn);
        }
      }
    } else {
      if (t >= 1) {  // attention on h_{t-1}
        const int job = (int)(blockIdx.x - 128) * 4 + w;  // 0..127
        const int mblk = job >> 5;                        // 0..3
        const int nstrip = job & 31;                      // 0..31
        const int m_base = mblk * 16;
        const int n_row = nstrip * 16 + l16;  // attn column / W_attn row

        v8f acc = {0.f, 0.f, 0.f, 0.f, 0.f, 0.f, 0.f, 0.f};
        const float* ap = hp + (size_t)(m_base + l16) * H_ + 2 * lh;
        const float* bp = W_attn + (size_t)n_row * H_ + 2 * lh;
#pragma unroll 8
        for (int k = 0; k < H_; k += 4) {
          acc = wmma4(*(const v2f*)(ap + k), *(const v2f*)(bp + k), acc);
        }
        const float bv = b_attn[n_row];
#pragma unroll
        for (int r = 0; r < 8; ++r) {
          int m = m_base + lh * 8 + r;  // batch index
          size_t idx = (size_t)m * H_ + n_row;
          float aval = tanhf(acc[r] + bv);
          float om = mbuf[idx];
          float nm = fmaxf(om, aval);
          float sc = __expf(om - nm);  // 0 when om == -inf
          float ev = __expf(aval - nm);
          sbuf[idx] = sbuf[idx] * sc + ev;
          wbuf[idx] = wbuf[idx] * sc + ev * hp[idx];
          mbuf[idx] = nm;
        }
      }
    }

    grid_barrier(cnt, gen);
  }

  // Finalize: context = wsum / denom, grid-strided over all 160 blocks.
  for (int i = (int)blockIdx.x * 128 + (int)threadIdx.x; i < B_ * H_;
       i += NBLK * 128) {
    out[i] = wbuf[i] / sbuf[i];
  }
}

extern "C" void kernel_launch(void* const* d_in, const int* in_sizes, int n_in,
                              void* d_out, int out_size, void* d_ws,
                              size_t ws_size, hipStream_t stream) {
  (void)in_sizes; (void)n_in; (void)out_size; (void)ws_size;
  const float* x = (const float*)d_in[0];
  const float* W_ih = (const float*)d_in[1];
  const float* W_hh = (const float*)d_in[2];
  const float* b_ih = (const float*)d_in[3];
  const float* b_hh = (const float*)d_in[4];
  const float* W_attn = (const float*)d_in[5];
  const float* b_attn = (const float*)d_in[6];

  float* ws = (float*)d_ws;
  const size_t N = (size_t)B_ * H_;  // 32768
  float* hb0 = ws;
  float* hb1 = ws + N;
  float* cb = ws + 2 * N;
  float* mb = ws + 3 * N;
  float* sb = ws + 4 * N;
  float* wb = ws + 5 * N;
  unsigned* bar = (unsigned*)(ws + 6 * N);

  lstm_init_kernel<<<(int)((N + 255) / 256), 256, 0, stream>>>(hb0, hb1, cb,
                                                               mb, sb, wb, bar);

  lstm_persist_kernel<<<NBLK, 128, 0, stream>>>(
      x, W_ih, W_hh, b_ih, b_hh, W_attn, b_attn, hb0, hb1, cb, mb, sb, wb,
      (float*)d_out, bar);
}